// MultiScaleROIExtractor_4818953306513
// MI455X (gfx1250) — compile-verified
//
#include <hip/hip_runtime.h>

typedef __attribute__((ext_vector_type(2))) float v2f;
typedef __attribute__((ext_vector_type(8))) float v8f;

#define IMG   960.0f
#define FH    240
#define FW    240
#define NCH   64
#define NBOX  4000

// ---------------------------------------------------------------------------
// ROI bilinear pooling: one block per (box, scale), lane == channel.
// Feature map (14.7MB) is L2-resident on MI455X (192MB L2), so the 4 gathers
// per sample point per channel are L2 hits after warmup.
// ---------------------------------------------------------------------------
__device__ __forceinline__ float roi_tap(const float* __restrict__ fmc, int x, int y) {
    bool valid = (x >= 0) & (x < FW) & (y >= 0) & (y < FH);
    int xc = min(max(x, 0), FW - 1);
    int yc = min(max(y, 0), FH - 1);
    float v = fmc[yc * FW + xc];
    return valid ? v : 0.0f;
}

__global__ void roi_pool_kernel(const float* __restrict__ fm,
                                const float* __restrict__ boxes,
                                float* __restrict__ pooled /* [3][NBOX][NCH] */) {
    const int n  = blockIdx.x;
    const int si = blockIdx.y;
    const int S  = (si == 0) ? 3 : (si == 1) ? 7 : 11;
    const int c  = threadIdx.x;          // 0..63 -> channel

    const float bx1 = boxes[n * 4 + 0];
    const float by1 = boxes[n * 4 + 1];
    const float bx2 = boxes[n * 4 + 2];
    const float by2 = boxes[n * 4 + 3];

    // normalize to [-1,1]
    const float x1 = bx1 / IMG * 2.0f - 1.0f;
    const float y1 = by1 / IMG * 2.0f - 1.0f;
    const float x2 = bx2 / IMG * 2.0f - 1.0f;
    const float y2 = by2 / IMG * 2.0f - 1.0f;
    const float cx = 0.5f * (x1 + x2);
    const float cy = 0.5f * (y1 + y2);
    const float w  = fmaxf(x2 - x1, 1e-6f);
    const float h  = fmaxf(y2 - y1, 1e-6f);

    const float* fmc = fm + c * (FH * FW);
    const float invS = 1.0f / (float)S;

    float acc = 0.0f;
    for (int i = 0; i < S; ++i) {
        const float basei = (2.0f * (float)i + 1.0f) * invS - 1.0f;
        const float gy = h * 0.5f * basei + cy;
        const float iy = ((gy + 1.0f) * (float)FH - 1.0f) * 0.5f;
        const float y0f = floorf(iy);
        const float dy  = iy - y0f;
        const int   y0  = (int)y0f;
        for (int j = 0; j < S; ++j) {
            const float basej = (2.0f * (float)j + 1.0f) * invS - 1.0f;
            const float gx = w * 0.5f * basej + cx;
            const float ix = ((gx + 1.0f) * (float)FW - 1.0f) * 0.5f;
            const float x0f = floorf(ix);
            const float dx  = ix - x0f;
            const int   x0  = (int)x0f;

            const float v00 = roi_tap(fmc, x0,     y0);
            const float v01 = roi_tap(fmc, x0 + 1, y0);
            const float v10 = roi_tap(fmc, x0,     y0 + 1);
            const float v11 = roi_tap(fmc, x0 + 1, y0 + 1);

            acc += v00 * (1.0f - dx) * (1.0f - dy)
                 + v01 * dx          * (1.0f - dy)
                 + v10 * (1.0f - dx) * dy
                 + v11 * dx          * dy;
        }
    }
    pooled[(si * NBOX + n) * NCH + c] = acc / (float)(S * S);
}

// ---------------------------------------------------------------------------
// Per-channel global mean of the feature map: one block per channel.
// ---------------------------------------------------------------------------
__global__ void global_mean_kernel(const float* __restrict__ fm, float* __restrict__ g) {
    __shared__ float sm[256];
    const int c = blockIdx.x;
    const float* p = fm + c * (FH * FW);
    float s = 0.0f;
    for (int i = threadIdx.x; i < FH * FW; i += 256) s += p[i];
    sm[threadIdx.x] = s;
    __syncthreads();
    for (int off = 128; off > 0; off >>= 1) {
        if (threadIdx.x < (unsigned)off) sm[threadIdx.x] += sm[threadIdx.x + off];
        __syncthreads();
    }
    if (threadIdx.x == 0) g[c] = sm[0] * (1.0f / (float)(FH * FW));
}

// ---------------------------------------------------------------------------
// Tiny MLP head on the (single) global feature vector: 64 -> 128 -> 64.
// ---------------------------------------------------------------------------
__global__ void global_head_kernel(const float* __restrict__ g,
                                   const float* __restrict__ W1, const float* __restrict__ b1,
                                   const float* __restrict__ W2, const float* __restrict__ b2,
                                   float* __restrict__ gfeat) {
    __shared__ float gs[64];
    __shared__ float h1[128];
    const int t = threadIdx.x;   // 0..127
    if (t < 64) gs[t] = g[t];
    __syncthreads();
    float s = b1[t];
    for (int c = 0; c < 64; ++c) s += gs[c] * W1[c * 128 + t];
    h1[t] = fmaxf(s, 0.0f);
    __syncthreads();
    if (t < 64) {
        float s2 = b2[t];
        for (int j = 0; j < 128; ++j) s2 += h1[j] * W2[j * 64 + t];
        gfeat[t] = fmaxf(s2, 0.0f);
    }
}

__global__ void broadcast_gfeat_kernel(const float* __restrict__ gfeat,
                                       float* __restrict__ cat /* [NBOX][256] */) {
    const int idx = blockIdx.x * blockDim.x + threadIdx.x;
    if (idx >= NBOX * NCH) return;
    const int n = idx >> 6;
    const int c = idx & 63;
    cat[n * 256 + 192 + c] = gfeat[c];
}

// ---------------------------------------------------------------------------
// Fused GEMM + bias + ReLU using V_WMMA_F32_16X16X4_F32 (full f32 precision).
// One wave per 16x16 output tile; K stepped by 4.
//
// A fragment (16x4, ISA 7.12.2): lane L holds row M = L%16; VGPR v holds
//   K = k0 + v + 2*(L/16).
// B fragment (4x16): lane L holds col N = L%16; VGPR v holds same K index.
// C/D (16x16 f32): VGPR r holds M = r + 8*(L/16), N = L%16.
// ---------------------------------------------------------------------------
__global__ void wmma_gemm_bias_relu(const float* __restrict__ A, int lda,
                                    const float* __restrict__ B, int ldb,
                                    const float* __restrict__ bias,
                                    float* __restrict__ C, int ldc, int c_col_off,
                                    int K) {
    const int lane = threadIdx.x;      // 0..31
    const int hf   = lane >> 4;        // half-wave select
    const int l16  = lane & 15;
    const int tm   = blockIdx.x * 16;  // output row tile
    const int tn   = blockIdx.y * 16;  // output col tile

    v8f acc = {};
    const float* arow = A + (size_t)(tm + l16) * lda;
    const float* bcol = B + tn + l16;

    for (int k0 = 0; k0 < K; k0 += 4) {
        const int ka = k0 + 2 * hf;
        v2f a, b;
        a.x = arow[ka];
        a.y = arow[ka + 1];
        b.x = bcol[(size_t)ka * ldb];
        b.y = bcol[(size_t)(ka + 1) * ldb];
        acc = __builtin_amdgcn_wmma_f32_16x16x4_f32(
            /*neg_a=*/false, a, /*neg_b=*/false, b,
            /*c_mod=*/(short)0, acc, /*reuse_a=*/false, /*reuse_b=*/false);
    }

    const float bv = bias[tn + l16];
#pragma unroll
    for (int r = 0; r < 8; ++r) {
        const int row = tm + r + 8 * hf;
        float v = acc[r] + bv;
        C[(size_t)row * ldc + c_col_off + tn + l16] = v > 0.0f ? v : 0.0f;
    }
}

// ---------------------------------------------------------------------------
// Host-side orchestration (graph-capture safe: launches only).
// ---------------------------------------------------------------------------
extern "C" void kernel_launch(void* const* d_in, const int* in_sizes, int n_in,
                              void* d_out, int out_size, void* d_ws, size_t ws_size,
                              hipStream_t stream) {
    const float* fm    = (const float*)d_in[0];
    const float* boxes = (const float*)d_in[1];
    const float* W1    = (const float*)d_in[2];
    const float* b1    = (const float*)d_in[3];
    const float* W2    = (const float*)d_in[4];
    const float* b2    = (const float*)d_in[5];
    const float* P1    = (const float*)d_in[6];
    const float* bp1   = (const float*)d_in[7];
    const float* P2    = (const float*)d_in[8];
    const float* bp2   = (const float*)d_in[9];
    float* out = (float*)d_out;
    float* ws  = (float*)d_ws;

    // Workspace layout (floats); everything is fully written before read.
    float* pooled = ws;                         // 3 * 4000 * 64 = 768000
    float* g      = pooled + 3 * NBOX * NCH;    // 64
    float* gfeat  = g + 64;                     // 64
    float* cat    = gfeat + 64;                 // 4000 * 256 = 1024000
    float* hidden = cat + NBOX * 256;           // 4000 * 128 = 512000

    // 1) ROI pooling for all 3 scales.
    roi_pool_kernel<<<dim3(NBOX, 3), 64, 0, stream>>>(fm, boxes, pooled);

    // 2) Global feature + its head (computed once, not per box).
    global_mean_kernel<<<64, 256, 0, stream>>>(fm, g);
    global_head_kernel<<<1, 128, 0, stream>>>(g, W1, b1, W2, b2, gfeat);
    broadcast_gfeat_kernel<<<(NBOX * NCH + 255) / 256, 256, 0, stream>>>(gfeat, cat);

    // 3) Per-scale heads (WMMA GEMMs), writing directly into cat columns.
    for (int si = 0; si < 3; ++si) {
        const float* X = pooled + (size_t)si * NBOX * NCH;
        // H = relu(X[4000,64] @ W1[64,128] + b1)
        wmma_gemm_bias_relu<<<dim3(NBOX / 16, 128 / 16), 32, 0, stream>>>(
            X, 64, W1, 128, b1, hidden, 128, 0, 64);
        // cat[:, si*64 : si*64+64] = relu(H @ W2[128,64] + b2)
        wmma_gemm_bias_relu<<<dim3(NBOX / 16, 64 / 16), 32, 0, stream>>>(
            hidden, 128, W2, 64, b2, cat, 256, si * 64, 128);
    }

    // 4) Final projection: relu(relu(cat @ P1 + bp1) @ P2 + bp2).
    wmma_gemm_bias_relu<<<dim3(NBOX / 16, 128 / 16), 32, 0, stream>>>(
        cat, 256, P1, 128, bp1, hidden, 128, 0, 256);
    wmma_gemm_bias_relu<<<dim3(NBOX / 16, 64 / 16), 32, 0, stream>>>(
        hidden, 128, P2, 64, bp2, out, 64, 0, 128);
}